// GCNConvWithTime_5858335392212
// MI455X (gfx1250) — compile-verified
//
#include <hip/hip_runtime.h>
#include <hip/hip_bf16.h>

typedef float v2f __attribute__((ext_vector_type(2)));
typedef float v8f __attribute__((ext_vector_type(8)));

#define IN_DIM  512
#define OUT_DIM 512
#define TC_DIM  256

// ---------------------------------------------------------------------------
// deg[i] = 1.0 (self loop)
__global__ void k_deg_init(float* __restrict__ deg, int N) {
    int i = blockIdx.x * blockDim.x + threadIdx.x;
    if (i < N) deg[i] = 1.0f;
}

// deg[dst[e]] += 1
__global__ void k_deg_accum(const int* __restrict__ ei, float* __restrict__ deg, int E) {
    int e = blockIdx.x * blockDim.x + threadIdx.x;
    if (e < E) atomicAdd(&deg[ei[E + e]], 1.0f);
}

// dinv[i] = rsqrt(deg[i])   (deg >= 1 always)
__global__ void k_dinv(const float* __restrict__ deg, float* __restrict__ dinv, int N) {
    int i = blockIdx.x * blockDim.x + threadIdx.x;
    if (i < N) dinv[i] = __frsqrt_rn(deg[i]);
}

// cvec[j] = b[j] + bt[j] + sum_k t_emb[k] * Wt[k, j]   (512 outputs)
__global__ void k_cvec(const float* __restrict__ t_emb, const float* __restrict__ Wt,
                       const float* __restrict__ b, const float* __restrict__ bt,
                       float* __restrict__ cvec) {
    int j = blockIdx.x * blockDim.x + threadIdx.x;
    if (j >= OUT_DIM) return;
    float s = b[j] + bt[j];
    #pragma unroll 4
    for (int k = 0; k < TC_DIM; ++k) s += t_emb[k] * Wt[k * OUT_DIM + j];
    cvec[j] = s;
}

// ---------------------------------------------------------------------------
// xw = x @ W using V_WMMA_F32_16X16X4_F32. One wave per 16x16 tile,
// K in steps of 4 (128 WMMA ops per tile).
__global__ void k_gemm_wmma(const float* __restrict__ x, const float* __restrict__ W,
                            float* __restrict__ xw, int N, int rowTiles) {
    const int wave = (blockIdx.x * blockDim.x + threadIdx.x) >> 5;
    const int lane = threadIdx.x & 31;
    const int colTile = wave & (OUT_DIM / 16 - 1);   // 32 col tiles
    const int rowTile = wave >> 5;
    if (rowTile >= rowTiles) return;

    const int row0 = rowTile * 16;
    const int col0 = colTile * 16;
    const int l15  = lane & 15;
    const int hi   = lane >> 4;

    int aRow = row0 + l15;
    if (aRow >= N) aRow = N - 1;                      // clamp (N%16==0 in practice)
    const float* __restrict__ aPtr = x + (size_t)aRow * IN_DIM + 2 * hi;
    const float* __restrict__ bPtr = W + (size_t)(2 * hi) * OUT_DIM + col0 + l15;

    v8f c = {};
    for (int k = 0; k < IN_DIM; k += 4) {
        v2f a = *(const v2f*)(aPtr + k);              // 2 consecutive K values -> b64 load
        v2f bm;
        bm.x = bPtr[(size_t)k * OUT_DIM];
        bm.y = bPtr[(size_t)(k + 1) * OUT_DIM];
        c = __builtin_amdgcn_wmma_f32_16x16x4_f32(
                false, a, false, bm, (short)0, c, false, false);
    }

    const int n     = col0 + l15;
    const int rbase = row0 + hi * 8;
    float* __restrict__ dPtr = xw + (size_t)rbase * OUT_DIM + n;
    if (row0 + 16 <= N) {                // WAVE-UNIFORM full-tile test -> scalar branch
        #pragma unroll
        for (int v = 0; v < 8; ++v) dPtr[(size_t)v * OUT_DIM] = c[v];
    } else {                             // ragged edge (unused for N=10000)
        for (int v = 0; v < 8; ++v)
            if (rbase + v < N) dPtr[(size_t)v * OUT_DIM] = c[v];
    }
}

// ---------------------------------------------------------------------------
// out[i,j] = cvec[j] + dinv[i]^2 * xw[i,j]  — float4 vectorized
__global__ void k_init_out(const float4* __restrict__ xw4, const float* __restrict__ dinv,
                           const float4* __restrict__ cvec4, float4* __restrict__ out4,
                           size_t total4) {
    size_t idx = (size_t)blockIdx.x * blockDim.x + threadIdx.x;
    if (idx >= total4) return;
    int i = (int)(idx >> 7);             // 128 float4 per row
    int q = (int)(idx & 127);
    float di = dinv[i];
    float s  = di * di;
    float4 xv = xw4[idx];
    float4 cv = cvec4[q];
    float4 r;
    r.x = cv.x + s * xv.x;
    r.y = cv.y + s * xv.y;
    r.z = cv.z + s * xv.z;
    r.w = cv.w + s * xv.w;
    out4[idx] = r;
}

// One wave per edge: out[dst, :] += dinv[src]*dinv[dst] * xw[src, :]
// 128 float4 per row -> 4 b128 loads + 16 f32 atomics per lane.
__global__ void k_edge_scatter(const int* __restrict__ ei, const float* __restrict__ xw,
                               const float* __restrict__ dinv, float* __restrict__ out,
                               int E) {
    int wave = (blockIdx.x * blockDim.x + threadIdx.x) >> 5;
    int lane = threadIdx.x & 31;
    if (wave >= E) return;
    int src = ei[wave];
    int dst = ei[E + wave];
    float norm = dinv[src] * dinv[dst];
    const float4* __restrict__ srow4 = (const float4*)(xw + (size_t)src * OUT_DIM);
    float* __restrict__ drow = out + (size_t)dst * OUT_DIM;
    #pragma unroll
    for (int c = 0; c < OUT_DIM / 128; ++c) {        // 4 chunks
        int q = lane + 32 * c;                       // float4 index (coalesced b128)
        float4 v = srow4[q];
        int col = 4 * q;
        atomicAdd(&drow[col + 0], norm * v.x);
        atomicAdd(&drow[col + 1], norm * v.y);
        atomicAdd(&drow[col + 2], norm * v.z);
        atomicAdd(&drow[col + 3], norm * v.w);
    }
}

// ---------------------------------------------------------------------------
extern "C" void kernel_launch(void* const* d_in, const int* in_sizes, int n_in,
                              void* d_out, int out_size, void* d_ws, size_t ws_size,
                              hipStream_t stream) {
    // setup_inputs order: x, t_emb, edge_index, W, b, Wt, bt
    const float* x     = (const float*)d_in[0];
    const float* t_emb = (const float*)d_in[1];
    const int*   ei    = (const int*)d_in[2];   // (2, E) int32 (JAX x64 off)
    const float* W     = (const float*)d_in[3];
    const float* b     = (const float*)d_in[4];
    const float* Wt    = (const float*)d_in[5];
    const float* bt    = (const float*)d_in[6];
    float*       out   = (float*)d_out;

    const int N = in_sizes[0] / IN_DIM;     // 10000
    const int E = in_sizes[2] / 2;          // 160000

    // Workspace layout (64B-aligned offsets)
    char* ws = (char*)d_ws;
    float* deg  = (float*)(ws + 0);                                        // N floats
    float* dinv = (float*)(ws + (((size_t)N * 4 + 63) & ~63ull));          // N floats
    float* cvec = (float*)((char*)dinv + (((size_t)N * 4 + 63) & ~63ull)); // 512 floats
    float* xw   = (float*)((char*)cvec + 4096);                            // N*512 floats

    // 1) degree (self-loop folded into init value of 1.0)
    k_deg_init <<<(N + 255) / 256, 256, 0, stream>>>(deg, N);
    k_deg_accum<<<(E + 255) / 256, 256, 0, stream>>>(ei, deg, E);
    k_dinv     <<<(N + 255) / 256, 256, 0, stream>>>(deg, dinv, N);

    // 2) per-column constant: b + t_emb @ Wt + bt
    k_cvec<<<(OUT_DIM + 255) / 256, 256, 0, stream>>>(t_emb, Wt, b, bt, cvec);

    // 3) xw = x @ W via FP32 WMMA (one wave per 16x16 tile, 8 waves/block)
    {
        int rowTiles = (N + 15) / 16;                 // 625
        int tiles    = rowTiles * (OUT_DIM / 16);     // 20000 waves
        int blocks   = (tiles + 7) / 8;
        k_gemm_wmma<<<blocks, 256, 0, stream>>>(x, W, xw, N, rowTiles);
    }

    // 4) out = cvec + self-loop message (fully overwrites poisoned d_out)
    {
        size_t total4 = (size_t)N * OUT_DIM / 4;
        int blocks = (int)((total4 + 255) / 256);
        k_init_out<<<blocks, 256, 0, stream>>>((const float4*)xw, dinv,
                                               (const float4*)cvec, (float4*)out, total4);
    }

    // 5) edge scatter-add (one wave per edge, L2-resident b128 gathers + atomics)
    {
        int blocks = (E + 7) / 8;                     // 8 waves (edges) per block
        k_edge_scatter<<<blocks, 256, 0, stream>>>(ei, xw, dinv, out, E);
    }
}